// FormulaNet_76484777607653
// MI455X (gfx1250) — compile-verified
//
#include <hip/hip_runtime.h>
#include <hip/hip_bf16.h>

typedef __attribute__((ext_vector_type(2))) float v2f;
typedef __attribute__((ext_vector_type(8))) float v8f;

#define N_GRAPHS 512
#define HIDDEN   128

// ---------------------------------------------------------------------------
// Generic fp32 WMMA GEMM:  C[M,128] = A[M,K] @ B[K,128] (+bias, optional relu)
// Block = 256 threads = 8 waves. Each block does a 16-row stripe; wave w owns
// the 16-column tile n0 = 16*w. A-stripe staged in LDS, shared by all waves.
// Uses V_WMMA_F32_16X16X4_F32 (exact fp32 semantics, ISA §7.12).
// Epilogue flags are template params so the hot loop/epilogue is branch-free.
// ---------------------------------------------------------------------------
template <int K, bool WITH_BIAS, bool DO_RELU>
__global__ __launch_bounds__(256) void gemm_wmma_f32(
    const float* __restrict__ A,   // [M, K] row-major
    const float* __restrict__ B,   // [K, 128] row-major
    const float* __restrict__ bias,// [128] (unused unless WITH_BIAS)
    float* __restrict__ C)         // [M, 128]
{
    __shared__ float As[16 * K];

    const int m0  = blockIdx.x * 16;
    const int tid = threadIdx.x;

    // cooperative stage of the 16xK A stripe (rows m0..m0+15 contiguous)
    for (int idx = tid; idx < 16 * K; idx += 256)
        As[idx] = A[(size_t)m0 * K + idx];
    __syncthreads();

    const int wave   = tid >> 5;      // 0..7 -> which 16-col tile
    const int lane   = tid & 31;
    const int n0     = wave * 16;
    const int lhalf  = lane >> 4;     // 0 or 1
    const int l15    = lane & 15;
    const int col    = n0 + l15;

    v8f c = {};
#pragma unroll
    for (int k = 0; k < K; k += 4) {
        // A 16x4 f32 fragment: lanes 0-15 -> K=k,k+1 ; lanes 16-31 -> K=k+2,k+3
        v2f a;
        const int ka = k + 2 * lhalf;
        a.x = As[l15 * K + ka];
        a.y = As[l15 * K + ka + 1];
        // B 4x16 fragment: VGPR0 rows {k,k+1}, VGPR1 rows {k+2,k+3} across halves
        v2f b;
        const int kb = k + lhalf;
        b.x = B[(size_t)kb * 128 + col];
        b.y = B[(size_t)(kb + 2) * 128 + col];
        // D = A*B + C   (8 args: neg_a, A, neg_b, B, c_mod, C, reuse_a, reuse_b)
        c = __builtin_amdgcn_wmma_f32_16x16x4_f32(false, a, false, b,
                                                  (short)0, c, false, false);
    }

    float bv = 0.0f;
    if (WITH_BIAS) bv = bias[col];    // one load per wave, outside the r-loop

#pragma unroll
    for (int r = 0; r < 8; ++r) {
        const int row = m0 + r + 8 * lhalf;   // C/D layout: VGPR r -> M=r / r+8
        float v = c[r];
        if (WITH_BIAS) v += bv;
        if (DO_RELU)   v = fmaxf(v, 0.0f);
        C[(size_t)row * 128 + col] = v;
    }
}

// ---------------------------------------------------------------------------
// Small helpers
// ---------------------------------------------------------------------------
__global__ void fill_zero(float* __restrict__ p, size_t n) {
    size_t i = (size_t)blockIdx.x * blockDim.x + threadIdx.x;
    if (i < n) p[i] = 0.0f;
}

__global__ void init_deg(float* __restrict__ deg, int n) {
    int i = blockIdx.x * blockDim.x + threadIdx.x;
    if (i < n) deg[i] = 1.0f;           // self-loop contributes 1 to deg[dst]
}

__global__ void deg_accum(const int* __restrict__ dst, float* __restrict__ deg, int e) {
    int i = blockIdx.x * blockDim.x + threadIdx.x;
    if (i < e) atomicAdd(&deg[dst[i]], 1.0f);
}

__global__ void finalize_dis(float* __restrict__ deg, int n) {
    int i = blockIdx.x * blockDim.x + threadIdx.x;
    if (i < n) deg[i] = rsqrtf(deg[i]); // deg >= 1 always (self-loops)
}

// h0 = relu(x @ W_d1 + b_d1),  x:[N,4], W:[4,4]
__global__ void dense1(const float* __restrict__ x, const float* __restrict__ W,
                       const float* __restrict__ b, float* __restrict__ h0, int n) {
    int i = blockIdx.x * blockDim.x + threadIdx.x;
    if (i >= n) return;
    float xi[4];
#pragma unroll
    for (int k = 0; k < 4; ++k) xi[k] = x[i * 4 + k];
#pragma unroll
    for (int j = 0; j < 4; ++j) {
        float acc = b[j];
#pragma unroll
        for (int k = 0; k < 4; ++k) acc += xi[k] * W[k * 4 + j];
        h0[i * 4 + j] = fmaxf(acc, 0.0f);
    }
}

// One wave per edge: acc[dst] += dis[src]*dis[dst] * hw[src]   (128 ch, fp32)
__global__ __launch_bounds__(256) void scatter_edges(
    const int* __restrict__ src, const int* __restrict__ dst,
    const float* __restrict__ dis, const float* __restrict__ hw,
    float* __restrict__ acc, int nEdges)
{
    int e = (int)(((size_t)blockIdx.x * blockDim.x + threadIdx.x) >> 5);
    int lane = threadIdx.x & 31;
    if (e >= nEdges) return;
    int s = src[e], d = dst[e];
    float norm = dis[s] * dis[d];
    const float4 v = ((const float4*)(hw + (size_t)s * 128))[lane];
    float* ap = acc + (size_t)d * 128 + lane * 4;
    atomicAdd(ap + 0, norm * v.x);
    atomicAdd(ap + 1, norm * v.y);
    atomicAdd(ap + 2, norm * v.z);
    atomicAdd(ap + 3, norm * v.w);
}

// h_out = relu(acc + dis[i]^2 * hw + bias)   (fused self-loop + bias + relu)
__global__ __launch_bounds__(256) void finish_layer(
    const float* __restrict__ acc, const float* __restrict__ hw,
    const float* __restrict__ dis, const float* __restrict__ b,
    float* __restrict__ hout, int n)
{
    size_t idx = (size_t)blockIdx.x * blockDim.x + threadIdx.x;
    if (idx >= (size_t)n * 128) return;
    int i = (int)(idx >> 7);
    int c = (int)(idx & 127);
    float dii = dis[i];
    float v = acc[idx] + dii * dii * hw[idx] + b[c];
    hout[idx] = fmaxf(v, 0.0f);
}

// Mean-pool accumulation: one wave per node
__global__ __launch_bounds__(256) void pool_accum(
    const float* __restrict__ h, const int* __restrict__ batch,
    float* __restrict__ sums, float* __restrict__ cnt, int n)
{
    int i = (int)(((size_t)blockIdx.x * blockDim.x + threadIdx.x) >> 5);
    int lane = threadIdx.x & 31;
    if (i >= n) return;
    int g = batch[i];
    const float4 v = ((const float4*)(h + (size_t)i * 128))[lane];
    float* sp = sums + (size_t)g * 128 + lane * 4;
    atomicAdd(sp + 0, v.x);
    atomicAdd(sp + 1, v.y);
    atomicAdd(sp + 2, v.z);
    atomicAdd(sp + 3, v.w);
    if (lane == 0) atomicAdd(&cnt[g], 1.0f);
}

__global__ void pool_div(float* __restrict__ sums, const float* __restrict__ cnt) {
    int idx = blockIdx.x * blockDim.x + threadIdx.x;
    if (idx >= N_GRAPHS * 128) return;
    int g = idx >> 7;
    sums[idx] /= fmaxf(cnt[g], 1.0f);
}

// ---------------------------------------------------------------------------
// Launcher
// ---------------------------------------------------------------------------
extern "C" void kernel_launch(void* const* d_in, const int* in_sizes, int n_in,
                              void* d_out, int out_size, void* d_ws, size_t ws_size,
                              hipStream_t stream) {
    // setup_inputs order:
    // 0:x 1:edge_index 2:batch 3:W_d1 4:b_d1 5:W_g1 6:b_g1 7:W_g2 8:b_g2
    // 9:W_g3 10:b_g3 11:W_d2 12:b_d2
    const float* x     = (const float*)d_in[0];
    const int*   eidx  = (const int*)d_in[1];
    const int*   batch = (const int*)d_in[2];
    const float* W_d1  = (const float*)d_in[3];
    const float* b_d1  = (const float*)d_in[4];
    const float* W_g1  = (const float*)d_in[5];
    const float* b_g1  = (const float*)d_in[6];
    const float* W_g2  = (const float*)d_in[7];
    const float* b_g2  = (const float*)d_in[8];
    const float* W_g3  = (const float*)d_in[9];
    const float* b_g3  = (const float*)d_in[10];
    const float* W_d2  = (const float*)d_in[11];
    const float* b_d2  = (const float*)d_in[12];

    const int N = in_sizes[0] / 4;      // 100000
    const int E = in_sizes[1] / 2;      // 1600000
    const int* src = eidx;              // edge_index[0]
    const int* dst = eidx + E;          // edge_index[1]

    // workspace carve-up (floats), 256B-aligned chunks
    float* ws = (float*)d_ws;
    size_t off = 0;
    auto alloc = [&](size_t nfloat) {
        float* p = ws + off;
        off += (nfloat + 63) & ~(size_t)63;
        return p;
    };
    float* dis  = alloc((size_t)N);            // deg -> dis (in place)
    float* h0   = alloc((size_t)N * 4);        // dense1 output
    float* h    = alloc((size_t)N * 128);      // layer activations
    float* hw   = alloc((size_t)N * 128);      // h @ W
    float* acc  = alloc((size_t)N * 128);      // edge-scatter accumulator
    float* sums = alloc((size_t)N_GRAPHS * 128);
    float* cnt  = alloc((size_t)N_GRAPHS);
    (void)ws_size; (void)n_in; (void)out_size;

    const size_t NH = (size_t)N * 128;
    const int gN     = (N + 255) / 256;
    const int gE     = (E + 255) / 256;
    const int gNH    = (int)((NH + 255) / 256);
    const int gEdgeW = (E + 7) / 8;            // 8 edges per 256-thread block
    const int gNodeW = (N + 7) / 8;

    // degrees -> dis = rsqrt(deg)  (self-loop folded into init)
    init_deg<<<gN, 256, 0, stream>>>(dis, N);
    deg_accum<<<gE, 256, 0, stream>>>(dst, dis, E);
    finalize_dis<<<gN, 256, 0, stream>>>(dis, N);

    // h0 = relu(x @ W_d1 + b_d1)
    dense1<<<gN, 256, 0, stream>>>(x, W_d1, b_d1, h0, N);

    // ---- GCN layer 1 (K=4) ----
    gemm_wmma_f32<4, false, false><<<N / 16, 256, 0, stream>>>(h0, W_g1, nullptr, hw);
    fill_zero<<<gNH, 256, 0, stream>>>(acc, NH);
    scatter_edges<<<gEdgeW, 256, 0, stream>>>(src, dst, dis, hw, acc, E);
    finish_layer<<<gNH, 256, 0, stream>>>(acc, hw, dis, b_g1, h, N);

    // ---- GCN layer 2 (K=128) ----
    gemm_wmma_f32<128, false, false><<<N / 16, 256, 0, stream>>>(h, W_g2, nullptr, hw);
    fill_zero<<<gNH, 256, 0, stream>>>(acc, NH);
    scatter_edges<<<gEdgeW, 256, 0, stream>>>(src, dst, dis, hw, acc, E);
    finish_layer<<<gNH, 256, 0, stream>>>(acc, hw, dis, b_g2, h, N);

    // ---- GCN layer 3 (K=128) ----
    gemm_wmma_f32<128, false, false><<<N / 16, 256, 0, stream>>>(h, W_g3, nullptr, hw);
    fill_zero<<<gNH, 256, 0, stream>>>(acc, NH);
    scatter_edges<<<gEdgeW, 256, 0, stream>>>(src, dst, dis, hw, acc, E);
    finish_layer<<<gNH, 256, 0, stream>>>(acc, hw, dis, b_g3, h, N);

    // ---- global mean pool ----
    fill_zero<<<(N_GRAPHS * 128 + 255) / 256, 256, 0, stream>>>(sums, N_GRAPHS * 128);
    fill_zero<<<(N_GRAPHS + 255) / 256, 256, 0, stream>>>(cnt, N_GRAPHS);
    pool_accum<<<gNodeW, 256, 0, stream>>>(h, batch, sums, cnt, N);
    pool_div<<<(N_GRAPHS * 128 + 255) / 256, 256, 0, stream>>>(sums, cnt);

    // ---- head: out = pooled @ W_d2 + b_d2 ----
    gemm_wmma_f32<128, true, false><<<N_GRAPHS / 16, 256, 0, stream>>>(
        sums, W_d2, b_d2, (float*)d_out);
}